// WaveletIIA_64415919505635
// MI455X (gfx1250) — compile-verified
//
#include <hip/hip_runtime.h>
#include <hip/hip_bf16.h>

// ---------------------------------------------------------------------------
// 2D DWT / attention-gate / IDWT pipeline for MI455X (gfx1250, wave32).
// FP32 end-to-end. Convolutions mapped onto V_WMMA_F32_16X16X4_F32 as banded
// matmuls. Workspace requirement: ~1.05 GiB (2*NAD + 4*NSB + feats/gates fp32).
// ---------------------------------------------------------------------------

typedef float v2f __attribute__((ext_vector_type(2)));
typedef float v8f __attribute__((ext_vector_type(8)));

#define BATCH 8
#define CH    64
#define BC    512              // BATCH*CH
#define HH_   512
#define WW_   512
#define NC    259              // (512 + 8 - 1) / 2
#define NAD_PER (HH_ * NC)     // per-(b,c) row-transformed plane: 132608
#define NSB_PER (NC * NC)      // per-(b,c) subband plane: 67081

static const size_t NADt = (size_t)BC * NAD_PER;   // 67,895,296 floats
static const size_t NSBt = (size_t)BC * NSB_PER;   // 34,345,472 floats

// Correlation taps: RL[t] = DEC_LO[7-t] (== REC_LO), RH[t] = DEC_HI[7-t] (== REC_HI).
// Forward DWT:  y[j]   = sum_t RL/RH[t] * ext[2j + t]
// Inverse DWT:  rec[j] = sum_k RL[j+6-2k]*cA[k] + RH[j+6-2k]*cD[k]
__device__ __constant__ float RL[8] = {
    0.23037781330885523f,  0.7148465705525415f,  0.6308807679295904f,
   -0.02798376941698385f, -0.18703481171888114f, 0.030841381835986965f,
    0.032883011666982945f,-0.010597401784997278f };
__device__ __constant__ float RH[8] = {
   -0.010597401784997278f,-0.032883011666982945f, 0.030841381835986965f,
    0.18703481171888114f, -0.02798376941698385f, -0.6308807679295904f,
    0.7148465705525415f,  -0.23037781330885523f };

__device__ inline v8f wmma_f32(v2f a, v2f b, v8f c) {
  // D(16x16,f32) = A(16x4,f32) x B(4x16,f32) + C
  return __builtin_amdgcn_wmma_f32_16x16x4_f32(false, a, false, b, (short)0, c,
                                               false, false);
}
__device__ inline int imin(int a, int b) { return a < b ? a : b; }

// ---------------------------------------------------------------------------
// Kernel 1: row DWT along W.  x(rows=BC*H, 512) -> a,d (rows, 259).
// A = data (M=row, K=ext col), B = banded filter (per-lane constants).
// One wave per WG; 16 rows per WG; 17 output tiles of 16 cols.
// ---------------------------------------------------------------------------
__global__ __launch_bounds__(32) void k_rowdwt(const float* __restrict__ X,
                                               float* __restrict__ A,
                                               float* __restrict__ D) {
  constexpr int ES = 561;                  // padded ext stride (bank-friendly)
  __shared__ float ext[16 * ES];
  const int lane = threadIdx.x;
  const int r0 = blockIdx.x * 16;

  for (int r = 0; r < 16; ++r) {
    const float* xr = X + (size_t)(r0 + r) * WW_;
    float* er = ext + r * ES;
    for (int c = lane; c < WW_; c += 32) er[6 + c] = xr[c];
    if (lane < 6) er[lane] = xr[5 - lane];            // symmetric left (6)
    if (lane < 7) er[518 + lane] = xr[511 - lane];    // symmetric right (7)
    for (int c = 525 + lane; c < ES; c += 32) er[c] = 0.f;
  }
  __syncthreads();

  const int M  = lane & 15;
  const int kb = (lane >> 4) * 2;          // 32-bit WMMA K-striping
  const int N  = lane & 15;

  v2f bl[10], bh[10];                      // B[u][N] = RL/RH[u - 2N]
#pragma unroll
  for (int c = 0; c < 10; ++c) {
    int s0 = 4 * c + kb - 2 * N, s1 = s0 + 1;
    bl[c].x = (s0 >= 0 && s0 < 8) ? RL[s0] : 0.f;
    bl[c].y = (s1 >= 0 && s1 < 8) ? RL[s1] : 0.f;
    bh[c].x = (s0 >= 0 && s0 < 8) ? RH[s0] : 0.f;
    bh[c].y = (s1 >= 0 && s1 < 8) ? RH[s1] : 0.f;
  }

  for (int j0 = 0; j0 < NC; j0 += 16) {
    v8f accL = {}; v8f accH = {};
#pragma unroll
    for (int c = 0; c < 10; ++c) {
      int q = 2 * j0 + 4 * c + kb;
      v2f a; a.x = ext[M * ES + q]; a.y = ext[M * ES + q + 1];
      accL = wmma_f32(a, bl[c], accL);
      accH = wmma_f32(a, bh[c], accH);
    }
    const int mb = 8 * (lane >> 4);
    const int col = j0 + N;
    if (col < NC) {
#pragma unroll
      for (int p = 0; p < 8; ++p) {
        size_t o = (size_t)(r0 + mb + p) * NC + col;
        A[o] = accL[p];
        D[o] = accH[p];
      }
    }
  }
}

// ---------------------------------------------------------------------------
// Kernel 2: col DWT along H. S(bc,512,259) -> OA,OB (bc,259,259).
// A = banded filter, B = data (K=ext h, N=w) so D-stores are w-coalesced.
// ---------------------------------------------------------------------------
__global__ __launch_bounds__(32) void k_coldwt(const float* __restrict__ S,
                                               float* __restrict__ OA,
                                               float* __restrict__ OB) {
  __shared__ float ext[560 * 16];
  const int lane = threadIdx.x;
  const int bc = blockIdx.x;
  const int w0 = blockIdx.y * 16;
  const float* Sb = S + (size_t)bc * NAD_PER;

  for (int idx = lane; idx < HH_ * 16; idx += 32) {
    int h = idx >> 4, wl = idx & 15;
    ext[(6 + h) * 16 + wl] = Sb[(size_t)h * NC + imin(w0 + wl, NC - 1)];
  }
  for (int idx = lane; idx < 6 * 16; idx += 32) {
    int i = idx >> 4, wl = idx & 15;
    ext[i * 16 + wl] = Sb[(size_t)(5 - i) * NC + imin(w0 + wl, NC - 1)];
  }
  for (int idx = lane; idx < 7 * 16; idx += 32) {
    int j = idx >> 4, wl = idx & 15;
    ext[(518 + j) * 16 + wl] = Sb[(size_t)(511 - j) * NC + imin(w0 + wl, NC - 1)];
  }
  for (int idx = lane; idx < (560 - 525) * 16; idx += 32)
    ext[(525 + (idx >> 4)) * 16 + (idx & 15)] = 0.f;
  __syncthreads();

  const int M  = lane & 15;
  const int kb = (lane >> 4) * 2;
  v2f al[10], ah[10];                      // A[M][u] = RL/RH[u - 2M]
#pragma unroll
  for (int c = 0; c < 10; ++c) {
    int s0 = 4 * c + kb - 2 * M, s1 = s0 + 1;
    al[c].x = (s0 >= 0 && s0 < 8) ? RL[s0] : 0.f;
    al[c].y = (s1 >= 0 && s1 < 8) ? RL[s1] : 0.f;
    ah[c].x = (s0 >= 0 && s0 < 8) ? RH[s0] : 0.f;
    ah[c].y = (s1 >= 0 && s1 < 8) ? RH[s1] : 0.f;
  }
  const int N = lane & 15;

  for (int j0 = 0; j0 < NC; j0 += 16) {
    v8f accL = {}; v8f accH = {};
#pragma unroll
    for (int c = 0; c < 10; ++c) {
      int q = 2 * j0 + 4 * c + kb;
      v2f b; b.x = ext[q * 16 + N]; b.y = ext[(q + 1) * 16 + N];
      accL = wmma_f32(al[c], b, accL);
      accH = wmma_f32(ah[c], b, accH);
    }
    const int hb = j0 + 8 * (lane >> 4);
    const int w = w0 + N;
    if (w < NC) {
#pragma unroll
      for (int p = 0; p < 8; ++p) {
        int hout = hb + p;
        if (hout < NC) {
          size_t o = (size_t)bc * NSB_PER + (size_t)hout * NC + w;
          OA[o] = accL[p];
          OB[o] = accH[p];
        }
      }
    }
  }
}

// ---------------------------------------------------------------------------
// Pooling: max/mean over last axis (LH case): wave-per-row shuffle butterfly.
// ---------------------------------------------------------------------------
__global__ void k_pool_lastaxis(const float* __restrict__ S, float* __restrict__ F) {
  const int bc = blockIdx.x;
  const int wv = threadIdx.x >> 5;
  const int lane = threadIdx.x & 31;
  const int h = blockIdx.y * 8 + wv;
  if (h >= NC) return;
  const float* row = S + (size_t)bc * NSB_PER + (size_t)h * NC;
  float mx = -3.402823466e38f, sm = 0.f;
  for (int w = lane; w < NC; w += 32) { float x = row[w]; mx = fmaxf(mx, x); sm += x; }
#pragma unroll
  for (int off = 16; off >= 1; off >>= 1) {
    mx = fmaxf(mx, __shfl_xor(mx, off, 32));
    sm += __shfl_xor(sm, off, 32);
  }
  if (lane == 0) {
    F[((size_t)bc * NC + h) * 2 + 0] = mx;
    F[((size_t)bc * NC + h) * 2 + 1] = sm * (1.f / NC);
  }
}

// Pooling over the h (middle) axis (HL case): thread-per-w, coalesced rows.
__global__ void k_pool_midaxis(const float* __restrict__ S, float* __restrict__ F) {
  const int bc = blockIdx.x;
  const int w = blockIdx.y * blockDim.x + threadIdx.x;
  if (w >= NC) return;
  const float* base = S + (size_t)bc * NSB_PER + w;
  float mx = -3.402823466e38f, sm = 0.f;
  for (int h = 0; h < NC; ++h) { float x = base[(size_t)h * NC]; mx = fmaxf(mx, x); sm += x; }
  F[((size_t)bc * NC + w) * 2 + 0] = mx;
  F[((size_t)bc * NC + w) * 2 + 1] = sm * (1.f / NC);
}

// Gate: y = w0*max + w1*mean + b; depthwise conv3 (zero pad); BN(eval); sigmoid.
__global__ void k_gate(const float* __restrict__ F,
                       const float* __restrict__ c1w, const float* __restrict__ c1b,
                       const float* __restrict__ c2w, const float* __restrict__ gam,
                       const float* __restrict__ bet, const float* __restrict__ rm,
                       const float* __restrict__ rv, float* __restrict__ G) {
  __shared__ float y[NC];
  const int bc = blockIdx.x;
  const int c = bc & (CH - 1);
  const float w0 = c1w[0], w1 = c1w[1], bb = c1b[0];
  for (int h = threadIdx.x; h < NC; h += blockDim.x) {
    size_t o = ((size_t)bc * NC + h) * 2;
    y[h] = w0 * F[o] + w1 * F[o + 1] + bb;
  }
  __syncthreads();
  const float t0 = c2w[c * 3 + 0], t1 = c2w[c * 3 + 1], t2 = c2w[c * 3 + 2];
  const float sc = rsqrtf(rv[c] + 1e-5f) * gam[c];
  const float sh = bet[c] - rm[c] * sc;
  for (int h = threadIdx.x; h < NC; h += blockDim.x) {
    float yl = (h > 0) ? y[h - 1] : 0.f;
    float yr = (h < NC - 1) ? y[h + 1] : 0.f;
    float z = t0 * yl + t1 * y[h] + t2 * yr;
    z = z * sc + sh;
    G[(size_t)bc * NC + h] = 1.f / (1.f + expf(-z));
  }
}

// ---------------------------------------------------------------------------
// Kernel 6: col IDWT along H with gate fused at LDS fill.
// rec[j][w] = sum_k RL[j+6-2k]*SA[k][w] + RH[j+6-2k]*SB[k][w]
// gmode==0: SB gated by G[k] (LH);  gmode==1: SA gated by G[w] (HL).
// A = banded inverse filter (indep. of tile since k0 = j0/2), B = coeff data.
// ---------------------------------------------------------------------------
__global__ __launch_bounds__(32) void k_colidwt(const float* __restrict__ SA,
                                                const float* __restrict__ SB,
                                                const float* __restrict__ G,
                                                float* __restrict__ OUT,
                                                int gmode) {
  __shared__ float As[264 * 16];
  __shared__ float Bs[264 * 16];
  const int lane = threadIdx.x;
  const int bc = blockIdx.x;
  const int w0 = blockIdx.y * 16;
  const float* gb  = G  + (size_t)bc * NC;
  const float* SAb = SA + (size_t)bc * NSB_PER;
  const float* SBb = SB + (size_t)bc * NSB_PER;

  for (int idx = lane; idx < NC * 16; idx += 32) {
    int k = idx >> 4, wl = idx & 15;
    int wc = imin(w0 + wl, NC - 1);
    float ga = (gmode == 1) ? gb[wc] : 1.f;
    float gk = (gmode == 0) ? gb[k]  : 1.f;
    As[k * 16 + wl] = SAb[(size_t)k * NC + wc] * ga;
    Bs[k * 16 + wl] = SBb[(size_t)k * NC + wc] * gk;
  }
  for (int idx = NC * 16 + lane; idx < 264 * 16; idx += 32) { As[idx] = 0.f; Bs[idx] = 0.f; }
  __syncthreads();

  const int M  = lane & 15;
  const int kb = (lane >> 4) * 2;
  v2f fl[3], fh[3];                        // A[M][u] = RL/RH[M + 6 - 2u]
#pragma unroll
  for (int c = 0; c < 3; ++c) {
    int s0 = M + 6 - 2 * (4 * c + kb), s1 = s0 - 2;
    fl[c].x = (s0 >= 0 && s0 < 8) ? RL[s0] : 0.f;
    fl[c].y = (s1 >= 0 && s1 < 8) ? RL[s1] : 0.f;
    fh[c].x = (s0 >= 0 && s0 < 8) ? RH[s0] : 0.f;
    fh[c].y = (s1 >= 0 && s1 < 8) ? RH[s1] : 0.f;
  }
  const int N = lane & 15;

  for (int jt = 0; jt < 32; ++jt) {
    const int j0 = jt * 16;
    const int k0 = j0 >> 1;
    v8f acc = {};
#pragma unroll
    for (int c = 0; c < 3; ++c) {
      int kk = k0 + 4 * c + kb;
      v2f ba; ba.x = As[kk * 16 + N]; ba.y = As[(kk + 1) * 16 + N];
      v2f bb; bb.x = Bs[kk * 16 + N]; bb.y = Bs[(kk + 1) * 16 + N];
      acc = wmma_f32(fl[c], ba, acc);
      acc = wmma_f32(fh[c], bb, acc);
    }
    const int jb = j0 + 8 * (lane >> 4);
    const int w = w0 + N;
    if (w < NC) {
#pragma unroll
      for (int p = 0; p < 8; ++p)
        OUT[(size_t)bc * NAD_PER + (size_t)(jb + p) * NC + w] = acc[p];
    }
  }
}

// ---------------------------------------------------------------------------
// Kernel 7: row IDWT along W + residual add.
// out[row][i] = x[row][i] + sum_k RL[i+6-2k]*a2[row][k] + RH[i+6-2k]*d2[row][k]
// A = coeff data (M=row), B = banded inverse filter.
// ---------------------------------------------------------------------------
__global__ __launch_bounds__(32) void k_rowidwt(const float* __restrict__ A2,
                                                const float* __restrict__ D2,
                                                const float* __restrict__ X,
                                                float* __restrict__ OUT) {
  constexpr int KS = 269;
  __shared__ float ae[16 * KS];
  __shared__ float de[16 * KS];
  const int lane = threadIdx.x;
  const int r0 = blockIdx.x * 16;

  for (int r = 0; r < 16; ++r) {
    const float* ar = A2 + (size_t)(r0 + r) * NC;
    const float* dr = D2 + (size_t)(r0 + r) * NC;
    for (int k = lane; k < NC; k += 32) { ae[r * KS + k] = ar[k]; de[r * KS + k] = dr[k]; }
    for (int k = NC + lane; k < KS; k += 32) { ae[r * KS + k] = 0.f; de[r * KS + k] = 0.f; }
  }
  __syncthreads();

  const int M  = lane & 15;
  const int kb = (lane >> 4) * 2;
  const int N  = lane & 15;
  v2f gl[3], gh[3];                        // B[u][N] = RL/RH[N + 6 - 2u]
#pragma unroll
  for (int c = 0; c < 3; ++c) {
    int s0 = N + 6 - 2 * (4 * c + kb), s1 = s0 - 2;
    gl[c].x = (s0 >= 0 && s0 < 8) ? RL[s0] : 0.f;
    gl[c].y = (s1 >= 0 && s1 < 8) ? RL[s1] : 0.f;
    gh[c].x = (s0 >= 0 && s0 < 8) ? RH[s0] : 0.f;
    gh[c].y = (s1 >= 0 && s1 < 8) ? RH[s1] : 0.f;
  }

  for (int it = 0; it < 32; ++it) {
    const int i0 = it * 16;
    const int k0 = i0 >> 1;
    v8f acc = {};
#pragma unroll
    for (int c = 0; c < 3; ++c) {
      int kk = k0 + 4 * c + kb;
      v2f aa; aa.x = ae[M * KS + kk]; aa.y = ae[M * KS + kk + 1];
      v2f dd; dd.x = de[M * KS + kk]; dd.y = de[M * KS + kk + 1];
      acc = wmma_f32(aa, gl[c], acc);
      acc = wmma_f32(dd, gh[c], acc);
    }
    const int mb = 8 * (lane >> 4);
    const int i = i0 + N;
#pragma unroll
    for (int p = 0; p < 8; ++p) {
      size_t o = (size_t)(r0 + mb + p) * WW_ + i;
      OUT[o] = X[o] + acc[p];
    }
  }
}

// ---------------------------------------------------------------------------
extern "C" void kernel_launch(void* const* d_in, const int* in_sizes, int n_in,
                              void* d_out, int out_size, void* d_ws, size_t ws_size,
                              hipStream_t stream) {
  (void)in_sizes; (void)n_in; (void)out_size; (void)ws_size;
  const float* x = (const float*)d_in[0];
  const float* lh_c1w = (const float*)d_in[1];
  const float* lh_c1b = (const float*)d_in[2];
  const float* lh_c2w = (const float*)d_in[3];
  const float* lh_g   = (const float*)d_in[4];
  const float* lh_b   = (const float*)d_in[5];
  const float* lh_m   = (const float*)d_in[6];
  const float* lh_v   = (const float*)d_in[7];
  const float* hl_c1w = (const float*)d_in[8];
  const float* hl_c1b = (const float*)d_in[9];
  const float* hl_c2w = (const float*)d_in[10];
  const float* hl_g   = (const float*)d_in[11];
  const float* hl_b   = (const float*)d_in[12];
  const float* hl_m   = (const float*)d_in[13];
  const float* hl_v   = (const float*)d_in[14];
  float* out = (float*)d_out;
  float* ws  = (float*)d_ws;

  // Workspace layout (floats). a/d are reused as a2/d2 after the subbands exist.
  const size_t o_a   = 0;
  const size_t o_d   = NADt;
  const size_t o_LL  = 2 * NADt;
  const size_t o_LH  = o_LL + NSBt;
  const size_t o_HL  = o_LL + 2 * NSBt;
  const size_t o_HH  = o_LL + 3 * NSBt;
  const size_t o_fLH = o_LL + 4 * NSBt;
  const size_t o_fHL = o_fLH + (size_t)BC * NC * 2;
  const size_t o_gLH = o_fHL + (size_t)BC * NC * 2;
  const size_t o_gHL = o_gLH + (size_t)BC * NC;

  // 1) row DWT: x -> a, d
  k_rowdwt<<<(BC * HH_) / 16, 32, 0, stream>>>(x, ws + o_a, ws + o_d);

  // 2) col DWT: a -> LL, LH ; d -> HL, HH
  k_coldwt<<<dim3(BC, 17), 32, 0, stream>>>(ws + o_a, ws + o_LL, ws + o_LH);
  k_coldwt<<<dim3(BC, 17), 32, 0, stream>>>(ws + o_d, ws + o_HL, ws + o_HH);

  // 3) pooled feats
  k_pool_lastaxis<<<dim3(BC, (NC + 7) / 8), 256, 0, stream>>>(ws + o_LH, ws + o_fLH);
  k_pool_midaxis<<<dim3(BC, (NC + 255) / 256), 256, 0, stream>>>(ws + o_HL, ws + o_fHL);

  // 4) gates
  k_gate<<<BC, 256, 0, stream>>>(ws + o_fLH, lh_c1w, lh_c1b, lh_c2w, lh_g, lh_b,
                                 lh_m, lh_v, ws + o_gLH);
  k_gate<<<BC, 256, 0, stream>>>(ws + o_fHL, hl_c1w, hl_c1b, hl_c2w, hl_g, hl_b,
                                 hl_m, hl_v, ws + o_gHL);

  // 5) col IDWT with fused gating: (LL, LH*gLH[h]) -> a2 ; (HL*gHL[w], HH) -> d2
  k_colidwt<<<dim3(BC, 17), 32, 0, stream>>>(ws + o_LL, ws + o_LH, ws + o_gLH,
                                             ws + o_a, 0);
  k_colidwt<<<dim3(BC, 17), 32, 0, stream>>>(ws + o_HL, ws + o_HH, ws + o_gHL,
                                             ws + o_d, 1);

  // 6) row IDWT + residual
  k_rowidwt<<<(BC * HH_) / 16, 32, 0, stream>>>(ws + o_a, ws + o_d, x, out);
}